// SVMLoss_49297634624085
// MI455X (gfx1250) — compile-verified
//
#include <hip/hip_runtime.h>
#include <hip/hip_bf16.h>

// Problem constants (fixed by the reference).
#define NN 8192
#define IT 256   // i-values per block (1 per thread)
#define JT 2048  // j-values per block tile

typedef __attribute__((ext_vector_type(2))) float v2f;
typedef __attribute__((ext_vector_type(8))) float v8f;
typedef __attribute__((ext_vector_type(4))) unsigned int v4u;
typedef __attribute__((ext_vector_type(4))) int v4i;
typedef __attribute__((ext_vector_type(8))) int v8i;

// ---------------------------------------------------------------------------
// Tensor Data Mover: 1D contiguous copy of `nelem` f32 from global -> LDS.
// D# per CDNA5 ISA:
//   group0: count=1, lds_addr, global_addr[56:0], type=2
//   group1: data_size=4B (code 2), tensor_dim0=nelem, tensor_dim1=1,
//           tile_dim0=nelem, tile_dim1=1, dim0_stride=nelem
// Issued once per wave (EXEC ignored by TDM); tracked with TENSORcnt.
// ---------------------------------------------------------------------------
__device__ __forceinline__ void tdm_load_1d(unsigned lds_off, const void* gptr,
                                            unsigned nelem) {
  unsigned long long ga = (unsigned long long)(uintptr_t)gptr;
  v4u g0;
  g0[0] = 1u;                                   // count=1, is_restore=0, gather off
  g0[1] = lds_off;                              // LDS byte address (wave-relative)
  g0[2] = (unsigned)(ga & 0xFFFFFFFFu);         // global_addr[31:0]
  g0[3] = (unsigned)((ga >> 32) & 0x01FFFFFFu)  // global_addr[56:32]
          | 0x80000000u;                        // type=2 ("image")
  v8i g1;
  g1[0] = 0x00020000;                           // workgroup_mask=0, data_size=2 (4B)
  g1[1] = (int)((nelem & 0xFFFFu) << 16);       // tensor_dim0[15:0] @ bits 63:48
  g1[2] = (int)((nelem >> 16) | (1u << 16));    // tensor_dim0[31:16], tensor_dim1=1
  g1[3] = (int)(nelem << 16);                   // tile_dim0 @ bits 127:112
  g1[4] = 1;                                    // tile_dim1=1, tile_dim2=0
  g1[5] = (int)nelem;                           // tensor_dim0_stride[31:0]
  g1[6] = 0;                                    // stride hi / dim1_stride lo
  g1[7] = 0;
  v4i z4 = {0, 0, 0, 0};
#if __clang_major__ >= 23
  v8i z8 = {0, 0, 0, 0, 0, 0, 0, 0};
  __builtin_amdgcn_tensor_load_to_lds(g0, g1, z4, z4, z8, 0);
#else
  __builtin_amdgcn_tensor_load_to_lds(g0, g1, z4, z4, 0);
#endif
}

// ---------------------------------------------------------------------------
// Sum 256 f32 values in LDS using V_WMMA_F32_16X16X4_F32 against a ones
// B-matrix: D = A x ones + C accumulates exact row sums (products by 1.0 and
// f32 accumulate are exact adds). 4 chained WMMAs cover 256 values; rows are
// folded with 7 adds + one half-wave shfl_xor. All lanes return the total.
// A 16x4 f32 layout (ISA §7.12.2): lane L<16 -> {A[L,0],A[L,1]},
// lane L>=16 -> {A[L-16,2],A[L-16,3]}.  A[m,k] := v[ch*64 + m*4 + k].
// ---------------------------------------------------------------------------
__device__ __forceinline__ float wmma_sum256(const float* v, int lane) {
#if __has_builtin(__builtin_amdgcn_wmma_f32_16x16x4_f32)
  v8f c = {0.f, 0.f, 0.f, 0.f, 0.f, 0.f, 0.f, 0.f};
  v2f b = {1.0f, 1.0f};
  const int m = lane & 15;
  const int kb = (lane >> 4) << 1;
  const float* base = v + m * 4 + kb;
#pragma unroll
  for (int ch = 0; ch < 4; ++ch) {
    v2f a;
    a[0] = base[ch * 64 + 0];
    a[1] = base[ch * 64 + 1];
    c = __builtin_amdgcn_wmma_f32_16x16x4_f32(false, a, false, b, (short)0, c,
                                              false, false);
  }
  // Lanes 0-15 hold rows 0..7 of D (any column), lanes 16-31 hold rows 8..15.
  float local = ((c[0] + c[1]) + (c[2] + c[3])) + ((c[4] + c[5]) + (c[6] + c[7]));
  return local + __shfl_xor(local, 16, 32);
#else
  float s = 0.f;
#pragma unroll
  for (int k = 0; k < 8; ++k) s += v[lane + 32 * k];
#pragma unroll
  for (int off = 16; off > 0; off >>= 1) s += __shfl_xor(s, off, 32);
  return s;
#endif
}

// ---------------------------------------------------------------------------
// Pair kernel: block (bx,by) handles i in [bx*256, bx*256+256) x
// j in [by*2048, by*2048+2048). j-tiles staged to LDS via TDM; inner loop is
// LDS-broadcast + VALU. Per-block partials (hinge2, count, regression) written
// deterministically to ws.
// ---------------------------------------------------------------------------
extern "C" __global__ void __launch_bounds__(256)
svm_pair(const float* __restrict__ pred, const float* __restrict__ target,
         float* __restrict__ ws) {
  __shared__ float p_t[JT];       // pred[j] tile
  __shared__ float tg_t[2 * JT];  // target rows (time,event interleaved)
  __shared__ float red_h[IT];
  __shared__ float red_c[IT];
  __shared__ float red_r[IT];

  const int tid = threadIdx.x;
  const int bx = blockIdx.x, by = blockIdx.y;
  const int i = bx * IT + tid;

  const float p_i = pred[i];
  const float t_i = target[2 * i];
  const float e_i = target[2 * i + 1];

  // One TDM issue per wave: wave 0 stages pred tile, wave 1 stages target tile.
  if (tid < 64) {
    if (tid < 32) {
      tdm_load_1d((unsigned)(uintptr_t)(void*)p_t, pred + (size_t)by * JT, JT);
    } else {
      tdm_load_1d((unsigned)(uintptr_t)(void*)tg_t, target + 2 * (size_t)by * JT,
                  2 * JT);
    }
    __builtin_amdgcn_s_wait_tensorcnt(0);
  }
  __syncthreads();

  const float a_i = p_i + 1.0f;  // hinge = max(a_i - p_j, 0)
  float h2 = 0.f, cc = 0.f;
#pragma unroll 4
  for (int jj = 0; jj < JT; ++jj) {
    const float pj = p_t[jj];
    const float tj = tg_t[2 * jj];
    const float mk = (tj > t_i) ? e_i : 0.0f;  // mask (0.0 or 1.0)
    const float h = fmaxf(a_i - pj, 0.0f);
    h2 = fmaf(mk * h, h, h2);
    cc += mk;
  }

  float rg = 0.f;
  if (by == 0) {  // regression term counted once per i
    float d = p_i - t_i;
    d = (e_i == 0.0f) ? fmaxf(d, 0.0f) : d;
    rg = d * d;
  }

  red_h[tid] = h2;
  red_c[tid] = cc;
  red_r[tid] = rg;
  __syncthreads();

  // All 8 waves run the WMMA reduction redundantly (EXEC all-1s everywhere).
  const int lane = tid & 31;
  const float bh = wmma_sum256(red_h, lane);
  const float bc = wmma_sum256(red_c, lane);
  const float br = wmma_sum256(red_r, lane);

  if (tid == 0) {
    const int blk = by * (NN / IT) + bx;  // gridDim.x == 32
    ws[blk] = bh;
    ws[128 + blk] = bc;
    if (by == 0) ws[256 + bx] = br;
  }
}

// ---------------------------------------------------------------------------
// Finalize: fixed-order f64 accumulation of the 128+128+32 partials.
// ---------------------------------------------------------------------------
extern "C" __global__ void svm_final(const float* __restrict__ ws,
                                     float* __restrict__ out) {
  if (threadIdx.x == 0 && blockIdx.x == 0) {
    double h = 0.0, c = 0.0, r = 0.0;
    for (int k = 0; k < 128; ++k) {
      h += (double)ws[k];
      c += (double)ws[128 + k];
    }
    for (int k = 0; k < 32; ++k) r += (double)ws[256 + k];
    const double cnt = (c < 1.0) ? 1.0 : c;
    out[0] = (float)(0.5 * (h / cnt) + 0.5 * (r / (double)NN));
  }
}

extern "C" void kernel_launch(void* const* d_in, const int* in_sizes, int n_in,
                              void* d_out, int out_size, void* d_ws,
                              size_t ws_size, hipStream_t stream) {
  const float* pred = (const float*)d_in[0];    // (N,1) f32
  const float* target = (const float*)d_in[1];  // (N,2) f32: [time,event]
  float* ws = (float*)d_ws;

  dim3 grid(NN / IT, NN / JT);  // 32 x 4 = 128 blocks
  svm_pair<<<grid, IT, 0, stream>>>(pred, target, ws);
  svm_final<<<1, 32, 0, stream>>>(ws, (float*)d_out);
}